// LocallyConnectedNetwork_58695023067964
// MI455X (gfx1250) — compile-verified
//
#include <hip/hip_runtime.h>

typedef float v2f __attribute__((ext_vector_type(2)));
typedef float v8f __attribute__((ext_vector_type(8)));

#define TILE  16     // images per block (== WMMA M)
#define H1STR 677    // 26*26=676, +1 pad -> odd stride, conflict-free across 64 banks
#define H2STR 577    // 24*24=576, +1 pad

// Fully fused: LC1 -> LC2 -> FC(WMMA f32 16x16x4, split-K over 8 waves).
__global__ __launch_bounds__(256)
void lcn_fused_kernel(const float* __restrict__ xg,
                      const float* __restrict__ w1g, const float* __restrict__ b1g,
                      const float* __restrict__ w2g, const float* __restrict__ b2g,
                      const float* __restrict__ fcw, const float* __restrict__ fcb,
                      float* __restrict__ out)
{
    __shared__ float h1[TILE * H1STR];   // 43,328 B : LC1 output
    __shared__ float h2[TILE * H2STR];   // 36,928 B : LC2 output (WMMA A source)
    __shared__ float fcp[16 * H2STR];    // 36,928 B : fc_w^T zero-padded to 16 cols (WMMA B source)
    __shared__ float red[8 * 256];       //  8,192 B : split-K partial C buffer

    const int tid     = threadIdx.x;
    const int imgBase = blockIdx.x * TILE;

    // ---- Stage B matrix into LDS (runs before first barrier, overlaps LC1) ----
    // fcp[n][k] = fc_w[n][k] for n<10, else 0.  Coalesced global, stride-1 LDS.
    for (int idx = tid; idx < 16 * 576; idx += 256) {
        const int n = idx / 576;
        const int k = idx - n * 576;
        fcp[n * H2STR + k] = (n < 10) ? fcw[n * 576 + k] : 0.0f;
    }

    if (tid == 0) {
        // warm L2 for the next tile's images (gfx1250 global_prefetch_b8)
        __builtin_prefetch(xg + (size_t)(imgBase + TILE) * 784, 0, 1);
    }

    // ---------------- Phase 1: LC1  x[28x28] -> h1[26x26] ----------------
    // idx = img*676 + px : consecutive lanes -> consecutive pixels of one
    // image -> coalesced global loads of x (L2-resident, 25.7 MB total).
    for (int idx = tid; idx < TILE * 676; idx += 256) {
        const int img = idx / 676;
        const int px  = idx - img * 676;
        const int y   = px / 26;
        const int xo  = px - y * 26;
        const float* xb = xg + (size_t)(imgBase + img) * 784;
        const float* wp = w1g + px * 9;
        float acc = b1g[px];
        #pragma unroll
        for (int i = 0; i < 3; ++i) {
            #pragma unroll
            for (int j = 0; j < 3; ++j)
                acc += xb[(y + i) * 28 + xo + j] * wp[i * 3 + j];
        }
        h1[img * H1STR + px] = acc;
    }
    __syncthreads();

    // ---------------- Phase 2: LC2  h1[26x26] -> h2[24x24] ----------------
    for (int idx = tid; idx < TILE * 576; idx += 256) {
        const int img = idx / 576;
        const int px  = idx - img * 576;
        const int y   = px / 24;
        const int xo  = px - y * 24;
        const float* hp = h1 + img * H1STR;
        const float* wp = w2g + px * 9;
        float acc = b2g[px];
        #pragma unroll
        for (int i = 0; i < 3; ++i) {
            #pragma unroll
            for (int j = 0; j < 3; ++j)
                acc += hp[(y + i) * 26 + xo + j] * wp[i * 3 + j];
        }
        h2[img * H2STR + px] = acc;
    }
    __syncthreads();   // h2 + fcp ready for WMMA

    // ---------------- Phase 3: FC via V_WMMA_F32_16X16X4_F32 ----------------
    // D[16 imgs x 16 cols] = h2[16 x 576] * fcp[576 x 16]
    // Split-K: wave w accumulates K in [w*72, w*72+72) -> 18 WMMAs per wave.
    // Fragment layouts (ISA 7.12.2, 32-bit, wave32):
    //   A 16x4 : lane = M (mod 16); lanes 0-15 carry K{0,1}, lanes 16-31 K{2,3}
    //   B 4x16 : lane = N (mod 16); lanes 0-15 carry K rows {0,1}, 16-31 {2,3}
    const int lane = tid & 31;
    const int wv   = tid >> 5;
    const int half = lane >> 4;
    const int mn   = lane & 15;

    v8f c;
    #pragma unroll
    for (int r = 0; r < 8; ++r) c[r] = 0.0f;

    const float* arow = h2  + mn * H2STR;   // odd stride -> bank-conflict-free
    const float* brow = fcp + mn * H2STR;

    const int k0 = wv * 72;
    #pragma unroll
    for (int kk = 0; kk < 18; ++kk) {
        const int ka = k0 + kk * 4 + half * 2;
        v2f a, b;
        a.x = arow[ka];
        a.y = arow[ka + 1];
        b.x = brow[ka];
        b.y = brow[ka + 1];
        // 8 args: (neg_a, A, neg_b, B, c_mod, C, reuse_a, reuse_b)
        c = __builtin_amdgcn_wmma_f32_16x16x4_f32(false, a, false, b,
                                                  (short)0, c, false, false);
    }

    // dump per-wave partial C (raw VGPR layout) into LDS for split-K reduce
    #pragma unroll
    for (int r = 0; r < 8; ++r)
        red[wv * 256 + r * 32 + lane] = c[r];
    __syncthreads();

    // Thread t owns C slot (vgpr r = t>>5, lane l = t&31):
    //   M = r + 8*(l>=16), N = l&15  (ISA 16x16 f32 C/D layout)
    {
        const int r = tid >> 5;
        const int l = tid & 31;
        float s = 0.0f;
        #pragma unroll
        for (int w = 0; w < 8; ++w) s += red[w * 256 + tid];
        const int M = r + ((l >> 4) << 3);
        const int N = l & 15;
        if (N < 10)
            out[(size_t)(imgBase + M) * 10 + N] = s + fcb[N];
    }
}

extern "C" void kernel_launch(void* const* d_in, const int* in_sizes, int n_in,
                              void* d_out, int out_size, void* d_ws, size_t ws_size,
                              hipStream_t stream) {
    const float* x   = (const float*)d_in[0];
    const float* w1  = (const float*)d_in[1];
    const float* b1  = (const float*)d_in[2];
    const float* w2  = (const float*)d_in[3];
    const float* b2  = (const float*)d_in[4];
    const float* fcw = (const float*)d_in[5];
    const float* fcb = (const float*)d_in[6];
    float* out = (float*)d_out;

    const int B    = in_sizes[0] / 784;       // 8192
    const int grid = (B + TILE - 1) / TILE;   // 512 blocks, 16 images each

    lcn_fused_kernel<<<grid, 256, 0, stream>>>(x, w1, b1, w2, b2, fcw, fcb, out);
}